// GCN_18038862643867
// MI455X (gfx1250) — compile-verified
//
#include <hip/hip_runtime.h>
#include <math.h>

typedef __attribute__((ext_vector_type(2))) float v2f;
typedef __attribute__((ext_vector_type(8))) float v8f;

#define NN 100000
#define NF 128
#define NC 40
#define NE 1600000
#define CHUNK 16   // k-steps (of K=4) staged in LDS at a time; 2 chunks cover K=128

// ---------------- degree / normalization ----------------
__global__ void deg_kernel(const int* __restrict__ dst, float* __restrict__ deg, int E) {
    int e = blockIdx.x * blockDim.x + threadIdx.x;
    if (e < E) atomicAdd(&deg[dst[e]], 1.0f);
}

__global__ void dinv_kernel(const float* __restrict__ deg, float* __restrict__ dinv,
                            float* __restrict__ sn, int N) {
    int i = blockIdx.x * blockDim.x + threadIdx.x;
    if (i < N) {
        float d = deg[i] + 1.0f;      // +1 self-loop
        float r = rsqrtf(d);
        dinv[i] = r;
        sn[i] = r * r;                // 1/deg
    }
}

__global__ void edge_norm_kernel(const int* __restrict__ src, const int* __restrict__ dst,
                                 const float* __restrict__ dinv, float* __restrict__ en, int E) {
    int e = blockIdx.x * blockDim.x + threadIdx.x;
    if (e < E) en[e] = dinv[src[e]] * dinv[dst[e]];
}

// ---------------- WMMA GEMM: C[N x OUTC] = A[N x 128] @ B[128 x OUTC] ----------------
// One wave -> 16 rows x (NT*16) cols, 8 waves/block -> 128 rows/block.
// B is staged per-block into LDS in per-lane WMMA fragment layout so each
// fragment is one conflict-free ds_load_b64 (lane-linear, 32 lanes x 8B = 64 banks).
template <int NT>
__global__ void __launch_bounds__(256)
gemm_wmma(const float* __restrict__ A, const float* __restrict__ B,
          const float* __restrict__ bias, float* __restrict__ C,
          int N, int OUTC, int relu) {
    __shared__ v2f Bs[CHUNK * NT * 32];   // NT=8 -> 32KB, NT=3 -> 12KB

    const int tid  = threadIdx.x;
    const int lane = tid & 31;
    const int wave = tid >> 5;
    const int base = blockIdx.x * 128 + wave * 16;
    const int half = lane >> 4;   // 0: K pair {kk,kk+1}, 1: {kk+2,kk+3}
    const int l16  = lane & 15;

    int arow = base + l16;
    if (arow > N - 1) arow = N - 1;          // clamp loads; stores masked below
    const float* aptr = A + (size_t)arow * NF + 2 * half;

    v8f acc[NT] = {};

    for (int chunk = 0; chunk < 32 / CHUNK; ++chunk) {
        const int s0 = chunk * CHUNK;

        // ---- cooperative fill of B fragments for this chunk ----
        if (chunk > 0) __syncthreads();      // previous chunk's reads done
#pragma unroll
        for (int fi = tid; fi < CHUNK * NT * 32; fi += 256) {
            const int fl   = fi & 31;
            const int ft   = (fi >> 5) % NT;
            const int fs   = fi / (32 * NT);
            const int fh   = fl >> 4;
            const int fcol = ft * 16 + (fl & 15);
            const int krow = 4 * (s0 + fs) + 2 * fh;
            v2f v;
            if (NT == 3) {                    // padded 40->48 cols for lin2
                v.x = (fcol < OUTC) ? B[(size_t)krow * OUTC + fcol] : 0.0f;
                v.y = (fcol < OUTC) ? B[(size_t)(krow + 1) * OUTC + fcol] : 0.0f;
            } else {
                v.x = B[(size_t)krow * OUTC + fcol];
                v.y = B[(size_t)(krow + 1) * OUTC + fcol];
            }
            Bs[fi] = v;
        }
        __syncthreads();

        if (chunk + 1 < 32 / CHUNK)           // prefetch next A chunk (global_prefetch_b8)
            __builtin_prefetch((const void*)(aptr + 4 * (s0 + CHUNK)), 0, 1);

        // ---- compute: 16 k-steps x NT wmma ----
#pragma unroll 4
        for (int sl = 0; sl < CHUNK; ++sl) {
            v2f a = *reinterpret_cast<const v2f*>(aptr + 4 * (s0 + sl));
            v2f bf[NT];
#pragma unroll
            for (int t = 0; t < NT; ++t)
                bf[t] = Bs[(sl * NT + t) * 32 + lane];   // ds_load_b64, conflict-free
#pragma unroll
            for (int t = 0; t < NT; ++t)
                acc[t] = __builtin_amdgcn_wmma_f32_16x16x4_f32(
                    false, a, false, bf[t], (short)0, acc[t], false, false);
        }
    }

    // D layout: vgpr r, lanes 0-15 -> M=r, lanes 16-31 -> M=8+r; N = l16
#pragma unroll
    for (int r = 0; r < 8; ++r) {
        const int outrow = base + r + 8 * half;
        if (outrow < N) {
#pragma unroll
            for (int t = 0; t < NT; ++t) {
                const int col = t * 16 + l16;
                if (NT != 3 || col < OUTC) {
                    float v = acc[t][r];
                    if (bias) v += bias[col];
                    if (relu) v = fmaxf(v, 0.0f);
                    C[(size_t)outrow * OUTC + col] = v;
                }
            }
        }
    }
}

// ---------------- edge gather * norm -> scatter-add (L2-resident atomics) ----------------
__global__ void edge_scatter(const float* __restrict__ h, const int* __restrict__ src,
                             const int* __restrict__ dst, const float* __restrict__ en,
                             float* __restrict__ agg, int E) {
    unsigned idx = blockIdx.x * blockDim.x + threadIdx.x;   // E * 32 threads, 4 feats each
    unsigned e = idx >> 5;
    if (e < (unsigned)E) {
        const int f = (idx & 31) << 2;
        const int s = src[e], d = dst[e];
        const float w = en[e];
        float4 m = *reinterpret_cast<const float4*>(h + (size_t)s * NF + f);
        float* ap = agg + (size_t)d * NF + f;
        atomicAdd(ap + 0, m.x * w);
        atomicAdd(ap + 1, m.y * w);
        atomicAdd(ap + 2, m.z * w);
        atomicAdd(ap + 3, m.w * w);
    }
}

// ---------------- out = relu(agg + h*self_norm + b) ----------------
__global__ void combine_relu(const float* __restrict__ agg, const float* __restrict__ h,
                             const float* __restrict__ sn, const float* __restrict__ bias,
                             float* __restrict__ out, int N) {
    int idx = blockIdx.x * blockDim.x + threadIdx.x;
    if (idx < N * NF) {
        const int node = idx >> 7;
        const int f = idx & (NF - 1);
        float v = agg[idx] + h[idx] * sn[node] + bias[f];
        out[idx] = fmaxf(v, 0.0f);
    }
}

// ---------------- log_softmax over 40 classes ----------------
__global__ void log_softmax_kernel(const float* __restrict__ logits, float* __restrict__ out, int N) {
    int i = blockIdx.x * blockDim.x + threadIdx.x;
    if (i < N) {
        const float* l = logits + (size_t)i * NC;
        float m = -INFINITY;
        for (int j = 0; j < NC; ++j) m = fmaxf(m, l[j]);
        float s = 0.0f;
        for (int j = 0; j < NC; ++j) s += expf(l[j] - m);
        const float lse = m + logf(s);
        float* o = out + (size_t)i * NC;
        for (int j = 0; j < NC; ++j) o[j] = l[j] - lse;
    }
}

// ---------------- driver ----------------
extern "C" void kernel_launch(void* const* d_in, const int* in_sizes, int n_in,
                              void* d_out, int out_size, void* d_ws, size_t ws_size,
                              hipStream_t stream) {
    const float* x    = (const float*)d_in[0];
    const int* ei     = (const int*)d_in[1];      // [2, E] int32
    const int* src    = ei;
    const int* dst    = ei + NE;
    const float* W1   = (const float*)d_in[2];
    const float* b1   = (const float*)d_in[3];
    const float* W2   = (const float*)d_in[4];
    const float* b2   = (const float*)d_in[5];
    const float* W3   = (const float*)d_in[6];
    const float* b3   = (const float*)d_in[7];
    const float* l1w  = (const float*)d_in[8];
    const float* l1b  = (const float*)d_in[9];
    const float* l2w  = (const float*)d_in[10];
    const float* l2b  = (const float*)d_in[11];
    float* out = (float*)d_out;

    float* ws   = (float*)d_ws;
    float* deg  = ws;                               // N
    float* dinv = deg + NN;                         // N
    float* sn   = dinv + NN;                        // N
    float* en   = sn + NN;                          // E
    float* bufA = en + NE;                          // N*128  (3N+E = 1.9M floats, 16B aligned)
    float* bufB = bufA + (size_t)NN * NF;           // N*128

    const int T = 256;
    const int gE   = (NE + T - 1) / T;
    const int gN   = (NN + T - 1) / T;
    const int gNF  = (NN * NF + T - 1) / T;
    const int gSc  = ((NE * 32) + T - 1) / T;
    const int gGemm = (NN + 127) / 128;

    // normalization precompute
    hipMemsetAsync(deg, 0, NN * sizeof(float), stream);
    deg_kernel<<<gE, T, 0, stream>>>(dst, deg, NE);
    dinv_kernel<<<gN, T, 0, stream>>>(deg, dinv, sn, NN);
    edge_norm_kernel<<<gE, T, 0, stream>>>(src, dst, dinv, en, NE);

    // GCN layer 1
    gemm_wmma<8><<<gGemm, T, 0, stream>>>(x, W1, nullptr, bufA, NN, NF, 0);
    hipMemsetAsync(bufB, 0, (size_t)NN * NF * sizeof(float), stream);
    edge_scatter<<<gSc, T, 0, stream>>>(bufA, src, dst, en, bufB, NE);
    combine_relu<<<gNF, T, 0, stream>>>(bufB, bufA, sn, b1, bufB, NN);

    // GCN layer 2
    gemm_wmma<8><<<gGemm, T, 0, stream>>>(bufB, W2, nullptr, bufA, NN, NF, 0);
    hipMemsetAsync(bufB, 0, (size_t)NN * NF * sizeof(float), stream);
    edge_scatter<<<gSc, T, 0, stream>>>(bufA, src, dst, en, bufB, NE);
    combine_relu<<<gNF, T, 0, stream>>>(bufB, bufA, sn, b2, bufB, NN);

    // GCN layer 3
    gemm_wmma<8><<<gGemm, T, 0, stream>>>(bufB, W3, nullptr, bufA, NN, NF, 0);
    hipMemsetAsync(bufB, 0, (size_t)NN * NF * sizeof(float), stream);
    edge_scatter<<<gSc, T, 0, stream>>>(bufA, src, dst, en, bufB, NE);
    combine_relu<<<gNF, T, 0, stream>>>(bufB, bufA, sn, b3, bufB, NN);

    // lin1 + relu
    gemm_wmma<8><<<gGemm, T, 0, stream>>>(bufB, l1w, l1b, bufA, NN, NF, 1);
    // lin2 (40 cols, padded to 48)
    gemm_wmma<3><<<gGemm, T, 0, stream>>>(bufA, l2w, l2b, bufB, NN, NC, 0);
    // log_softmax
    log_softmax_kernel<<<gN, T, 0, stream>>>(bufB, out, NN);
}